// Grid_4097398800632
// MI455X (gfx1250) — compile-verified
//
#include <hip/hip_runtime.h>

// 2D hash-grid encoding (Instant-NGP style), bilinear interpolation.
// Memory-bound gather kernel: table (134 MB) is kept L2-resident by using
// non-temporal hints on all streaming traffic (x in, out), while the 4
// random 32B row gathers per point use default (temporal) policy.

typedef float f32x2 __attribute__((ext_vector_type(2)));
typedef float f32x4 __attribute__((ext_vector_type(4)));

#define HASH_MASK  ((1u << 22) - 1u)   // HASHMAP_SIZE = 2^22 (power of two -> mask)
#define PRIME0     73856093u
#define PRIME1     19349663u

__global__ __launch_bounds__(256) void hashgrid_encode_kernel(
    const f32x2* __restrict__ x,       // (N, 2) points, streamed once
    const f32x4* __restrict__ table4,  // (2^22, 8) f32 = (2^23) float4 rows-halves
    f32x4* __restrict__ out4,          // (N, 8) f32 = (2N) float4
    int n_points)
{
    int i = blockIdx.x * blockDim.x + threadIdx.x;
    if (i >= n_points) return;

    // Streaming input: non-temporal (do not pollute L2, table lives there).
    f32x2 p = __builtin_nontemporal_load(&x[i]);

    // Reference rounding order: (x + 1.0) * 0.5 * 1024.0  (last two muls exact)
    float sx = (p.x + 1.0f) * 0.5f * 1024.0f;
    float sy = (p.y + 1.0f) * 0.5f * 1024.0f;
    int ixi = (int)sx;                 // sx >= 0 -> trunc == floor == astype(int64)
    int iyi = (int)sy;
    float fx = sx - (float)ixi;
    float fy = sy - (float)iyi;

    // 32-bit hash is exact modulo 2^22 (low multiply bits are width-invariant).
    unsigned a0 = (unsigned)ixi * PRIME0;
    unsigned a1 = a0 + PRIME0;         // (ix+1)*P0
    unsigned b0 = (unsigned)iyi * PRIME1;
    unsigned b1 = b0 + PRIME1;         // (iy+1)*P1

    unsigned h0 = (a0 ^ b0) & HASH_MASK;   // corner (ix  , iy  )  w = (1-fx)(1-fy)
    unsigned h1 = (a1 ^ b0) & HASH_MASK;   // corner (ix+1, iy  )  w =  fx  (1-fy)
    unsigned h2 = (a0 ^ b1) & HASH_MASK;   // corner (ix  , iy+1)  w = (1-fx) fy
    unsigned h3 = (a1 ^ b1) & HASH_MASK;   // corner (ix+1, iy+1)  w =  fx   fy

    float gx = 1.0f - fx, gy = 1.0f - fy;
    float w0 = gx * gy, w1 = fx * gy, w2 = gx * fy, w3 = fx * fy;

    // Issue all 8 gather loads up front (one s_wait ladder, max MLP).
    // Each table row is 32 B = two b128 loads; temporal policy -> L2 resident.
    size_t r0 = (size_t)h0 * 2u, r1 = (size_t)h1 * 2u,
           r2 = (size_t)h2 * 2u, r3 = (size_t)h3 * 2u;
    f32x4 t0a = table4[r0],     t0b = table4[r0 + 1];
    f32x4 t1a = table4[r1],     t1b = table4[r1 + 1];
    f32x4 t2a = table4[r2],     t2b = table4[r2 + 1];
    f32x4 t3a = table4[r3],     t3b = table4[r3 + 1];

    f32x4 accA = t0a * w0;
    f32x4 accB = t0b * w0;
    accA += t1a * w1;  accB += t1b * w1;
    accA += t2a * w2;  accB += t2b * w2;
    accA += t3a * w3;  accB += t3b * w3;

    // Streaming output: non-temporal b128 stores.
    __builtin_nontemporal_store(accA, &out4[(size_t)i * 2u]);
    __builtin_nontemporal_store(accB, &out4[(size_t)i * 2u + 1u]);
}

extern "C" void kernel_launch(void* const* d_in, const int* in_sizes, int n_in,
                              void* d_out, int out_size, void* d_ws, size_t ws_size,
                              hipStream_t stream) {
    const f32x2* x      = (const f32x2*)d_in[0];   // (N_POINTS, 2) f32
    const f32x4* table4 = (const f32x4*)d_in[1];   // (HASHMAP_SIZE, 8) f32
    f32x4*       out4   = (f32x4*)d_out;           // (N_POINTS, 8) f32

    int n_points = in_sizes[0] / 2;                // 4,194,304
    int block = 256;
    int grid  = (n_points + block - 1) / block;    // 16384 blocks

    hashgrid_encode_kernel<<<grid, block, 0, stream>>>(x, table4, out4, n_points);
}